// FaceClassifier_6906307412161
// MI455X (gfx1250) — compile-verified
//
#include <hip/hip_runtime.h>
#include <math.h>

// FaceClassifier for MI455X (gfx1250, wave32).
// centers -> knn(top-16 d2) -> mutual-dedupe mask -> 3x triconv (f32 WMMA 16x16x4) -> logits -> softmax.

#define NK   16
#define HID  64

typedef __attribute__((ext_vector_type(2))) float v2f;
typedef __attribute__((ext_vector_type(8))) float v8f;

// ---------------------------------------------------------------- centers
__global__ void centers_kernel(const float* __restrict__ pos, float* __restrict__ pos_c, int n) {
  int i = blockIdx.x * blockDim.x + threadIdx.x;
  if (i >= n) return;
  const float inv3 = 1.0f / 3.0f;
#pragma unroll
  for (int c = 0; c < 3; ++c)
    pos_c[i * 3 + c] = (pos[i * 9 + c] + pos[i * 9 + 3 + c] + pos[i * 9 + 6 + c]) * inv3;
}

// ---------------------------------------------------------------- knn (one wave32 per row)
#define KNN_TILE  1024
#define KNN_WAVES 8
__global__ __launch_bounds__(256) void knn_kernel(const float* __restrict__ pos_c,
                                                  int* __restrict__ knn_out, int n) {
  __shared__ float sx[KNN_TILE], sy[KNN_TILE], sz[KNN_TILE];
  __shared__ float s_md[KNN_WAVES][NK * 32];
  __shared__ int   s_mi[KNN_WAVES][NK * 32];

  int tid = threadIdx.x;
  int lane = tid & 31, w = tid >> 5;
  int row = blockIdx.x * KNN_WAVES + w;
  bool active = row < n;

  float px = 0.f, py = 0.f, pz = 0.f;
  if (active) { px = pos_c[row * 3]; py = pos_c[row * 3 + 1]; pz = pos_c[row * 3 + 2]; }

  float d2l[NK]; int idl[NK];
#pragma unroll
  for (int s = 0; s < NK; ++s) { d2l[s] = INFINITY; idl[s] = 0; }

  int ntiles = (n + KNN_TILE - 1) / KNN_TILE;
  for (int t = 0; t < ntiles; ++t) {
    int base = t * KNN_TILE;
    for (int idx = tid; idx < KNN_TILE; idx += 256) {
      int g = base + idx;
      if (g < n) { sx[idx] = pos_c[g * 3]; sy[idx] = pos_c[g * 3 + 1]; sz[idx] = pos_c[g * 3 + 2]; }
      else       { sx[idx] = INFINITY;     sy[idx] = 0.f;              sz[idx] = 0.f; }
    }
    __syncthreads();
    if (active) {
      for (int j = lane; j < KNN_TILE; j += 32) {
        int g = base + j;
        float dx = sx[j] - px, dy = sy[j] - py, dz = sz[j] - pz;
        float d2 = fmaf(dx, dx, fmaf(dy, dy, dz * dz));
        if (g == row || g >= n) d2 = INFINITY;
        if (d2 < d2l[NK - 1]) {            // common path: 1 compare, no insert
          float cd = d2; int ci = g;
#pragma unroll
          for (int s = 0; s < NK; ++s) {   // sorted bubble-insert, registers only
            if (cd < d2l[s]) { float td = d2l[s]; int ti = idl[s];
                               d2l[s] = cd; idl[s] = ci; cd = td; ci = ti; }
          }
        }
      }
    }
    __syncthreads();
  }
  // merge 32 lanes x 16 -> 16 via LDS argmin rounds + wave32 shuffle reduce
#pragma unroll
  for (int s = 0; s < NK; ++s) { s_md[w][lane * NK + s] = d2l[s]; s_mi[w][lane * NK + s] = idl[s]; }
  __syncthreads();
  for (int r = 0; r < NK; ++r) {
    float bv = INFINITY; int bp = 0;
#pragma unroll
    for (int s = 0; s < NK; ++s) {
      float v = s_md[w][lane * NK + s];
      if (v < bv) { bv = v; bp = lane * NK + s; }
    }
    for (int off = 16; off; off >>= 1) {
      float ov = __shfl_xor(bv, off);
      int   op = __shfl_xor(bp, off);
      if (ov < bv) { bv = ov; bp = op; }
    }
    if (lane == 0) {
      if (active) knn_out[row * NK + r] = s_mi[w][bp];
      s_md[w][bp] = INFINITY;
    }
    __syncthreads();
  }
}

// ---------------------------------------------------------------- dedupe mask (pass-2 edge valid iff NOT mutual)
__global__ void valid2_kernel(const int* __restrict__ knn, unsigned char* __restrict__ vmask, int n) {
  int t = blockIdx.x * blockDim.x + threadIdx.x;
  if (t >= n * NK) return;
  int i = t >> 4;
  int a = knn[t];
  unsigned char v = 1;
#pragma unroll
  for (int j = 0; j < NK; ++j)
    if (knn[a * NK + j] == i) v = 0;
  vmask[t] = v;
}

// ---------------------------------------------------------------- utility
__global__ void zero_kernel(float* __restrict__ p, int count) {
  int i = blockIdx.x * blockDim.x + threadIdx.x;
  if (i < count) p[i] = 0.0f;
}

// ---------------------------------------------------------------- triconv: one 16-edge tile per wave, f32 WMMA
// Tile t < n  : pass-1 edges (knn_j(t) -> t), all valid, uniform dst = t.
// Tile t >= n : pass-2 edges (t-n -> knn_j(t-n)), masked by vmask (mutual dedupe).
// Weights live in LDS K-pair-interleaved so each lane's B operand is a single aligned
// ds_load_b64 into an even VGPR pair; rows padded to 80 v2f so the hi lane-half
// (K row +1) sits exactly +32 banks away -> conflict-free across the wave.
template <int F, bool RELU_IN>
__global__ __launch_bounds__(128) void triconv_kernel(
    const float* __restrict__ xin, const float* __restrict__ pos_c,
    const int* __restrict__ knn, const unsigned char* __restrict__ vmask,
    const float* __restrict__ w1, const float* __restrict__ b1,
    const float* __restrict__ w2, const float* __restrict__ b2,
    float* __restrict__ out, int n) {
  constexpr int DIN  = F + 3;
  constexpr int KP   = (DIN + 3) & ~3;  // pad K to multiple of 4 (12 or 68)
  constexpr int KP2  = KP / 2;
  constexpr int WPAD = 80;              // (80*2) % 64 banks == 32 -> halves disjoint
  constexpr int WAVES = 4;

  __shared__ v2f   s_w1p[KP2][WPAD];
  __shared__ v2f   s_w2p[HID / 2][WPAD];
  __shared__ float s_b1[HID];
  __shared__ float s_b2[HID];
  __shared__ float s_a[WAVES][16][68];  // A rows (16 x KP), reused for H (16 x 64)
  __shared__ int   s_src[WAVES][16];
  __shared__ int   s_dst[WAVES][16];
  __shared__ unsigned char s_vld[WAVES][16];

  int tid = threadIdx.x;
  int lane = tid & 31, w = tid >> 5;

  // stage weights, K-pair interleaved
  for (int idx = tid; idx < KP2 * HID; idx += 128) {
    int kk = idx / HID, c = idx % HID;
    int k0 = 2 * kk, k1 = 2 * kk + 1;
    v2f v;
    v.x = (k0 < DIN) ? w1[k0 * HID + c] : 0.0f;
    v.y = (k1 < DIN) ? w1[k1 * HID + c] : 0.0f;
    s_w1p[kk][c] = v;
  }
  for (int idx = tid; idx < (HID / 2) * HID; idx += 128) {
    int kk = idx / HID, c = idx % HID;
    v2f v;
    v.x = w2[(2 * kk) * HID + c];
    v.y = w2[(2 * kk + 1) * HID + c];
    s_w2p[kk][c] = v;
  }
  if (tid < HID) { s_b1[tid] = b1[tid]; s_b2[tid] = b2[tid]; }

  int tile = blockIdx.x * WAVES + w;
  bool tvalid = tile < 2 * n;
  if (tvalid && lane < 16) {
    int e = tile * 16 + lane;
    int src, dst; unsigned char v;
    if (tile < n) { dst = tile; src = knn[e]; v = 1; }
    else { int e2 = e - n * NK; src = tile - n; dst = knn[e2]; v = vmask[e2]; }
    s_src[w][lane] = src; s_dst[w][lane] = dst; s_vld[w][lane] = v;
  }
  __syncthreads();

  if (tvalid) {  // stage A = [relu?(x_src) | rel_pos | 0-pad]
    for (int idx = lane; idx < 16 * KP; idx += 32) {
      int r = idx / KP, c = idx % KP;
      int src = s_src[w][r], dst = s_dst[w][r];
      float v;
      if (c < F) {
        v = xin[src * F + c];
        if (RELU_IN) v = fmaxf(v, 0.0f);   // fused relu of previous layer
      } else if (c < DIN) {
        v = pos_c[src * 3 + (c - F)] - pos_c[dst * 3 + (c - F)];
      } else {
        v = 0.0f;
      }
      s_a[w][r][c] = v;
    }
  }
  __syncthreads();

  int m  = lane & 15;       // row (A) / col (B,D) within 16
  int hi = lane >> 4;       // K-half select per 16x4 f32 A/B layout
  v8f zv = {0.f, 0.f, 0.f, 0.f, 0.f, 0.f, 0.f, 0.f};
  v8f acc[4] = {zv, zv, zv, zv};

  // GEMM1: H = A(16xKP) * W1(KPx64)
#pragma unroll
  for (int kt = 0; kt < KP / 4; ++kt) {
    int kb = 4 * kt + 2 * hi;
    v2f a = *(const v2f*)&s_a[w][m][kb];        // single b64, 8B aligned
#pragma unroll
    for (int nt = 0; nt < 4; ++nt) {
      v2f b = s_w1p[2 * kt + hi][nt * 16 + m];  // single b64, conflict-free
      acc[nt] = __builtin_amdgcn_wmma_f32_16x16x4_f32(false, a, false, b,
                                                      (short)0, acc[nt], false, false);
    }
  }
  __syncthreads();  // A dead; rewrite s_a with H = relu(acc + b1)
  if (tvalid) {
#pragma unroll
    for (int nt = 0; nt < 4; ++nt) {
      int col = nt * 16 + m;
#pragma unroll
      for (int v = 0; v < 8; ++v)
        s_a[w][v + 8 * hi][col] = fmaxf(acc[nt][v] + s_b1[col], 0.0f);
    }
  }
  __syncthreads();

  v8f acc2[4] = {zv, zv, zv, zv};
  // GEMM2: O = H(16x64) * W2(64x64)
#pragma unroll
  for (int kt = 0; kt < HID / 4; ++kt) {
    int kb = 4 * kt + 2 * hi;
    v2f a = *(const v2f*)&s_a[w][m][kb];
#pragma unroll
    for (int nt = 0; nt < 4; ++nt) {
      v2f b = s_w2p[2 * kt + hi][nt * 16 + m];
      acc2[nt] = __builtin_amdgcn_wmma_f32_16x16x4_f32(false, a, false, b,
                                                       (short)0, acc2[nt], false, false);
    }
  }

  if (tvalid) {
    if (tile < n) {  // pass-1: uniform dst, pre-reduce 8 rows -> 1 atomic per (lane,nt)
      int dst = tile;
#pragma unroll
      for (int nt = 0; nt < 4; ++nt) {
        int col = nt * 16 + m;
        float sum = 8.0f * s_b2[col];   // 8 valid edges per lane-half contribute b2
#pragma unroll
        for (int v = 0; v < 8; ++v) sum += acc2[nt][v];
        atomicAdd(&out[dst * HID + col], sum);
      }
    } else {         // pass-2: scattered dst, per-row masked atomics
#pragma unroll
      for (int nt = 0; nt < 4; ++nt) {
        int col = nt * 16 + m;
#pragma unroll
        for (int v = 0; v < 8; ++v) {
          int r = v + 8 * hi;
          if (s_vld[w][r])
            atomicAdd(&out[s_dst[w][r] * HID + col], acc2[nt][v] + s_b2[col]);
        }
      }
    }
  }
}

// ---------------------------------------------------------------- head (fused relu on input)
__global__ void logits_kernel(const float* __restrict__ x, const float* __restrict__ w_out,
                              const float* __restrict__ b_out, float* __restrict__ logits, int n) {
  int i = blockIdx.x * blockDim.x + threadIdx.x;
  if (i >= n) return;
  float s = b_out[0];
#pragma unroll
  for (int c = 0; c < HID; ++c) s += fmaxf(x[i * HID + c], 0.0f) * w_out[c];
  logits[i] = s;
}

__global__ __launch_bounds__(1024) void softmax_kernel(const float* __restrict__ logits,
                                                       float* __restrict__ out, int n) {
  __shared__ float red[32];
  int tid = threadIdx.x;
  float mx = -INFINITY;
  for (int i = tid; i < n; i += 1024) mx = fmaxf(mx, logits[i]);
  for (int off = 16; off; off >>= 1) mx = fmaxf(mx, __shfl_xor(mx, off));
  if ((tid & 31) == 0) red[tid >> 5] = mx;
  __syncthreads();
  if (tid < 32) {
    float v = red[tid];
    for (int off = 16; off; off >>= 1) v = fmaxf(v, __shfl_xor(v, off));
    red[tid] = v;
  }
  __syncthreads();
  float gmax = red[0];
  __syncthreads();
  float s = 0.0f;
  for (int i = tid; i < n; i += 1024) s += expf(logits[i] - gmax);
  for (int off = 16; off; off >>= 1) s += __shfl_xor(s, off);
  if ((tid & 31) == 0) red[tid >> 5] = s;
  __syncthreads();
  if (tid < 32) {
    float v = red[tid];
    for (int off = 16; off; off >>= 1) v += __shfl_xor(v, off);
    red[tid] = v;
  }
  __syncthreads();
  float inv = 1.0f / red[0];
  for (int i = tid; i < n; i += 1024) out[i] = expf(logits[i] - gmax) * inv;
}

// ---------------------------------------------------------------- launch
extern "C" void kernel_launch(void* const* d_in, const int* in_sizes, int n_in,
                              void* d_out, int out_size, void* d_ws, size_t ws_size,
                              hipStream_t stream) {
  (void)n_in; (void)out_size; (void)ws_size;
  const float* pos   = (const float*)d_in[0];
  const float* probs = (const float*)d_in[1];
  const float* w1a[3] = {(const float*)d_in[2], (const float*)d_in[6],  (const float*)d_in[10]};
  const float* b1a[3] = {(const float*)d_in[3], (const float*)d_in[7],  (const float*)d_in[11]};
  const float* w2a[3] = {(const float*)d_in[4], (const float*)d_in[8],  (const float*)d_in[12]};
  const float* b2a[3] = {(const float*)d_in[5], (const float*)d_in[9],  (const float*)d_in[13]};
  const float* w_out = (const float*)d_in[14];
  const float* b_out = (const float*)d_in[15];
  int n = in_sizes[0] / 9;

  char* ws = (char*)d_ws;
  size_t off = 0;
  auto alloc = [&](size_t bytes) -> void* {
    void* p = ws + off;
    off = (off + bytes + 255) & ~(size_t)255;
    return p;
  };
  float* pos_c      = (float*)alloc((size_t)n * 3 * sizeof(float));
  int*   knn        = (int*)alloc((size_t)n * NK * sizeof(int));
  unsigned char* vm = (unsigned char*)alloc((size_t)n * NK);
  float* xa         = (float*)alloc((size_t)n * HID * sizeof(float));
  float* xb         = (float*)alloc((size_t)n * HID * sizeof(float));
  float* lg         = (float*)alloc((size_t)n * sizeof(float));

  centers_kernel<<<(n + 255) / 256, 256, 0, stream>>>(pos, pos_c, n);
  knn_kernel<<<(n + KNN_WAVES - 1) / KNN_WAVES, 256, 0, stream>>>(pos_c, knn, n);
  valid2_kernel<<<(n * NK + 255) / 256, 256, 0, stream>>>(knn, vm, n);

  int tcblocks = (2 * n + 3) / 4;  // 16-edge tiles, 4 waves/block
  int feat = n * HID;

  zero_kernel<<<(feat + 255) / 256, 256, 0, stream>>>(xa, feat);
  triconv_kernel<8, false><<<tcblocks, 128, 0, stream>>>(probs, pos_c, knn, vm,
                                                         w1a[0], b1a[0], w2a[0], b2a[0], xa, n);

  zero_kernel<<<(feat + 255) / 256, 256, 0, stream>>>(xb, feat);
  triconv_kernel<64, true><<<tcblocks, 128, 0, stream>>>(xa, pos_c, knn, vm,
                                                         w1a[1], b1a[1], w2a[1], b2a[1], xb, n);

  zero_kernel<<<(feat + 255) / 256, 256, 0, stream>>>(xa, feat);
  triconv_kernel<64, true><<<tcblocks, 128, 0, stream>>>(xb, pos_c, knn, vm,
                                                         w1a[2], b1a[2], w2a[2], b2a[2], xa, n);

  logits_kernel<<<(n + 255) / 256, 256, 0, stream>>>(xa, w_out, b_out, lg, n);
  softmax_kernel<<<1, 1024, 0, stream>>>(lg, (float*)d_out, n);
}